// Rot3dImg_21440476741673
// MI455X (gfx1250) — compile-verified
//
#include <hip/hip_runtime.h>
#include <math.h>
#include <stdint.h>

#ifndef M_PI
#define M_PI 3.14159265358979323846
#endif

#define NSIDE 256
#define SLICE (NSIDE * NSIDE)          // 65536 elements = 256 KB fp32 per z-slice

typedef unsigned int tdm_v4u __attribute__((ext_vector_type(4)));
typedef int          tdm_v8i __attribute__((ext_vector_type(8)));
typedef int          tdm_v4i __attribute__((ext_vector_type(4)));

// One block per z-slice: TDM-stage the 256KB slice into LDS, then each of the
// 1024 threads bilinearly samples 64 output pixels from LDS and stores float4s.
__global__ __launch_bounds__(1024, 1)
void rot3d_slice_kernel(const float* __restrict__ x, float* __restrict__ out,
                        float c, float s) {
    __shared__ float tile[SLICE];      // 256 KB of the 320 KB/WGP LDS

    const int k = blockIdx.x;          // z-slice index (iz == k exactly, wz == 0)
    const int t = threadIdx.x;
    const float* src = x + (size_t)k * SLICE;

#if defined(__gfx1250__) && __has_builtin(__builtin_amdgcn_tensor_load_to_lds) && __has_builtin(__builtin_amdgcn_s_wait_tensorcnt)
    if (t < 32) {                      // single wave issues the DMA once
        uint64_t ga = (uint64_t)(uintptr_t)src;
        uint32_t la = (uint32_t)(uintptr_t)&tile[0];   // low 32 bits = LDS byte offset
        // D# group 0: [1:0]=count=1, [63:32]=lds_addr, [120:64]=global_addr, [127:126]=type=2
        tdm_v4u g0 = { 1u, la, (uint32_t)ga, (uint32_t)(ga >> 32) | (2u << 30) };
        // D# group 1:
        //  w0: workgroup_mask=0, data_size=2 (4B) at [17:16]
        //  w1: tensor_dim0[15:0]=256 at bits [63:48]
        //  w2: tensor_dim0[31:16]=0, tensor_dim1[15:0]=256 at bits [95:80]
        //  w3: tensor_dim1[31:16]=0, tile_dim0=256 at bits [127:112]
        //  w4: tile_dim1=256 at [143:128], tile_dim2=0
        //  w5: tensor_dim0_stride[31:0]=256
        //  w6: tensor_dim0_stride[47:32]=0, tensor_dim1_stride[15:0]=0
        //  w7: tensor_dim1_stride[47:16]=1  (stride1 = 65536, unused for 2D tile)
        tdm_v8i g1 = { (int)(2u << 16),
                       (int)(256u << 16),
                       (int)(256u << 16),
                       (int)(256u << 16),
                       256, 256, 0, 1 };
        tdm_v4i g2 = { 0, 0, 0, 0 };   // tile_dim3/iterate unused
        tdm_v4i g3 = { 0, 0, 0, 0 };   // tile_dim4 unused
        tdm_v8i gx = { 0, 0, 0, 0, 0, 0, 0, 0 };  // extra operand (clang-23 6-arg form)
        __builtin_amdgcn_tensor_load_to_lds(g0, g1, g2, g3, gx, 0);
        __builtin_amdgcn_s_wait_tensorcnt(0);
    }
#else
    for (int idx = t; idx < SLICE; idx += 1024) tile[idx] = src[idx];
#endif
    __syncthreads();

    const float invn = 1.0f / (float)NSIDE;
    for (int base = 0; base < SLICE; base += 4096) {
        int p  = base + t * 4;         // 4 consecutive x-pixels per thread
        int j  = p >> 8;               // output row (H)
        int i0 = p & (NSIDE - 1);      // output col (W)
        float ysv = (2.0f * (float)j + 1.0f) * invn - 1.0f;
        float rr[4];
#pragma unroll
        for (int q = 0; q < 4; ++q) {
            int i = i0 + q;
            float xsv = (2.0f * (float)i + 1.0f) * invn - 1.0f;
            float gx = c * xsv - s * ysv;              // rotation about z
            float gy = s * xsv + c * ysv;
            float fx = ((gx + 1.0f) * 256.0f - 1.0f) * 0.5f;  // unnormalize
            float fy = ((gy + 1.0f) * 256.0f - 1.0f) * 0.5f;
            float fx0 = floorf(fx);
            float fy0 = floorf(fy);
            float wx = fx - fx0;
            float wy = fy - fy0;
            int ix0 = (int)fx0, iy0 = (int)fy0;
            int ix1 = ix0 + 1,  iy1 = iy0 + 1;
            float mx0 = (ix0 >= 0 && ix0 < NSIDE) ? 1.0f : 0.0f;
            float mx1 = (ix1 >= 0 && ix1 < NSIDE) ? 1.0f : 0.0f;
            float my0 = (iy0 >= 0 && iy0 < NSIDE) ? 1.0f : 0.0f;
            float my1 = (iy1 >= 0 && iy1 < NSIDE) ? 1.0f : 0.0f;
            int xc0 = min(max(ix0, 0), NSIDE - 1);
            int xc1 = min(max(ix1, 0), NSIDE - 1);
            int yc0 = min(max(iy0, 0), NSIDE - 1);
            int yc1 = min(max(iy1, 0), NSIDE - 1);
            float v00 = mx0 * my0 * tile[yc0 * NSIDE + xc0];
            float v01 = mx1 * my0 * tile[yc0 * NSIDE + xc1];
            float v10 = mx0 * my1 * tile[yc1 * NSIDE + xc0];
            float v11 = mx1 * my1 * tile[yc1 * NSIDE + xc1];
            rr[q] = (1.0f - wy) * ((1.0f - wx) * v00 + wx * v01)
                  +         wy  * ((1.0f - wx) * v10 + wx * v11);
        }
        float4 r;
        r.x = rr[0]; r.y = rr[1]; r.z = rr[2]; r.w = rr[3];
        *(float4*)(out + (size_t)k * SLICE + p) = r;   // coalesced b128 store
    }

    // theta = [[c,-s,0,0],[s,c,0,0],[0,0,1,0]] appended after the volume
    if (k == 0 && t < 12) {
        float tv = 0.0f;
        if (t == 0 || t == 5)      tv = c;
        else if (t == 1)           tv = -s;
        else if (t == 4)           tv = s;
        else if (t == 10)          tv = 1.0f;
        out[(size_t)NSIDE * SLICE + t] = tv;
    }
}

extern "C" void kernel_launch(void* const* d_in, const int* in_sizes, int n_in,
                              void* d_out, int out_size, void* d_ws, size_t ws_size,
                              hipStream_t stream) {
    (void)in_sizes; (void)n_in; (void)d_ws; (void)ws_size; (void)out_size;
    const float* x = (const float*)d_in[0];
    float* out = (float*)d_out;
    // angle = START_VIEW * pi/180 + 0 * pi/VIEWS = 45 degrees (double, like numpy)
    double angle = 45.0 * (M_PI / 180.0);
    float c = (float)cos(angle);
    float s = (float)sin(angle);
    rot3d_slice_kernel<<<dim3(NSIDE), dim3(1024), 0, stream>>>(x, out, c, s);
}